// create_72395968741976
// MI455X (gfx1250) — compile-verified
//
#include <hip/hip_runtime.h>
#include <hip/hip_bf16.h>
#include <stdint.h>

// LDPC normalized-min-sum BP decoder for MI455X (gfx1250).
// B=64 batches, M=8192 checks, N=16384 variables, DEG=8, 50 iterations.
// Internal layouts put batch (64 = 2 waves) innermost for full coalescing.
// CDNA5 features: async global->LDS double-buffered msg prefetch (ASYNCcnt).

constexpr int Bc   = 64;
constexpr int Mc   = 8192;
constexpr int Nc   = 16384;
constexpr int DEGc = 8;
constexpr int ITERS = 50;
constexpr float ALPHAc = 0.75f;

constexpr int TPB   = 256;            // 8 waves (wave32)
constexpr int ROWS  = 4;              // check rows per block step
constexpr int TILE  = ROWS * DEGc * Bc; // 2048 floats = 8 KB per buffer
constexpr int NSTEPS = Mc / ROWS;     // 2048
constexpr int CHK_BLOCKS = 512;       // persistent blocks -> 4 steps each (pipeline depth)

// ---- CDNA5 async global->LDS copy + wait (inline asm; no confirmed builtin) ----
#if defined(__has_builtin) && __has_builtin(__builtin_amdgcn_s_wait_asynccnt)
#define WAIT_ASYNCCNT(n) __builtin_amdgcn_s_wait_asynccnt(n)
#else
#define WAIT_ASYNCCNT(n) asm volatile("s_wait_asynccnt %0" :: "i"(n) : "memory")
#endif

// Copies 32 bytes per lane (2 x b128): tile bytes [goff, goff+16) and [goff+4096, ...)
// from (base + offsets) into LDS at lds_addr (+0 / +4096). IOFFSET applies to both
// the LDS and global addresses per the CDNA5 ISA async pseudocode.
__device__ __forceinline__ void async_tile_load(unsigned lds_addr, unsigned goff,
                                                const float* base) {
  asm volatile(
      "global_load_async_to_lds_b128 %0, %1, %2\n\t"
      "global_load_async_to_lds_b128 %0, %1, %2 offset:4096"
      :: "v"(lds_addr), "v"(goff), "s"(base) : "memory");
}

// ---------------- setup kernels ----------------

// U[v*64+b] = llr0[b*N+v] for v<N, +inf for v==N   (i over (N+1)*64)
__global__ __launch_bounds__(TPB) void k_build_u(const float* __restrict__ llr0,
                                                 float* __restrict__ U) {
  int i = blockIdx.x * TPB + threadIdx.x;
  if (i >= (Nc + 1) * Bc) return;
  int b = i & 63, v = i >> 6;
  U[i] = (v < Nc) ? llr0[(size_t)b * Nc + v] : __builtin_inff();
}

// S[m*64+b] = 1 - 2*synd[b*M+m]
__global__ __launch_bounds__(TPB) void k_sgns(const int* __restrict__ synd,
                                              float* __restrict__ S) {
  int i = blockIdx.x * TPB + threadIdx.x;
  if (i >= Mc * Bc) return;
  int b = i & 63, m = i >> 6;
  S[i] = 1.0f - 2.0f * (float)synd[(size_t)b * Mc + m];
}

__global__ __launch_bounds__(TPB) void k_zero_i32(int* __restrict__ p, int n) {
  int i = blockIdx.x * TPB + threadIdx.x;
  if (i < n) p[i] = 0;
}

// histogram of non-dummy edge columns (edge e = m*DEG+d)
__global__ __launch_bounds__(TPB) void k_count(const int* __restrict__ vcol,
                                               int* __restrict__ counts) {
  int e = blockIdx.x * TPB + threadIdx.x;
  if (e >= Mc * DEGc) return;
  int c = vcol[e];
  if (c < Nc) atomicAdd(&counts[c], 1);
}

// exclusive scan of counts[0..Nc] -> rowptr[0..Nc+1]; single block of 256 threads.
__global__ __launch_bounds__(TPB) void k_scan(const int* __restrict__ counts,
                                              int* __restrict__ rowptr) {
  __shared__ int part[TPB];
  __shared__ int partscan[TPB];
  const int NBINS = Nc + 1;            // 16385
  const int PER = (NBINS + TPB - 1) / TPB; // 65
  int tid = threadIdx.x;
  int base = tid * PER;
  int sum = 0;
  for (int j = 0; j < PER; ++j) {
    int idx = base + j;
    if (idx < NBINS) sum += counts[idx];
  }
  part[tid] = sum;
  __syncthreads();
  if (tid == 0) {
    int acc = 0;
    for (int t = 0; t < TPB; ++t) { partscan[t] = acc; acc += part[t]; }
  }
  __syncthreads();
  int acc = partscan[tid];
  for (int j = 0; j < PER; ++j) {
    int idx = base + j;
    if (idx < NBINS) { rowptr[idx] = acc; acc += counts[idx]; }
  }
  if (tid == TPB - 1) rowptr[NBINS] = acc;
}

__global__ __launch_bounds__(TPB) void k_cursor(const int* __restrict__ rowptr,
                                                int* __restrict__ cursor) {
  int i = blockIdx.x * TPB + threadIdx.x;
  if (i <= Nc) cursor[i] = rowptr[i];
}

__global__ __launch_bounds__(TPB) void k_fill(const int* __restrict__ vcol,
                                              int* __restrict__ cursor,
                                              int* __restrict__ edges) {
  int e = blockIdx.x * TPB + threadIdx.x;
  if (e >= Mc * DEGc) return;
  int c = vcol[e];
  if (c < Nc) {
    int p = atomicAdd(&cursor[c], 1);
    edges[p] = e;
  }
}

// per-column insertion sort by edge id -> deterministic CSR regardless of atomic order
__global__ __launch_bounds__(TPB) void k_sort(const int* __restrict__ rowptr,
                                              int* __restrict__ edges) {
  int v = blockIdx.x * TPB + threadIdx.x;
  if (v > Nc) return;
  int s = rowptr[v], e = rowptr[v + 1];
  for (int i = s + 1; i < e; ++i) {
    int key = edges[i];
    int j = i - 1;
    while (j >= s && edges[j] > key) { edges[j + 1] = edges[j]; --j; }
    edges[j + 1] = key;
  }
}

// ---------------- per-iteration kernels ----------------

// Check-node update: one thread per (check m, batch b). Extrinsic sign-parity and
// two-min kept in registers. msg tile for each row-group is prefetched into LDS
// with CDNA5 async global->LDS copies, double buffered across the persistent loop.
__global__ __launch_bounds__(TPB) void k_check(const float* __restrict__ lv,
                                               float* __restrict__ msg,
                                               const float* __restrict__ sgnS,
                                               const int* __restrict__ vcol,
                                               int first) {
  __shared__ __align__(16) float smsg[2][TILE];
  const int tid = threadIdx.x;
  const int b = tid & 63;
  const int mi = tid >> 6;
  const int stride = gridDim.x;

  int rg = blockIdx.x;
  if (rg >= NSTEPS) return;

  if (!first) {
    unsigned lds0 = (unsigned)(size_t)(&smsg[0][0]) + (unsigned)(tid * 16);
    unsigned g0 = (unsigned)(rg * (TILE * 4)) + (unsigned)(tid * 16);
    async_tile_load(lds0, g0, msg);
  }
  int buf = 0;
  for (; rg < NSTEPS; rg += stride) {
    const int m = rg * ROWS + mi;
    if (!first) {
      int nrg = rg + stride;
      if (nrg < NSTEPS) {
        unsigned ldsn = (unsigned)(size_t)(&smsg[buf ^ 1][0]) + (unsigned)(tid * 16);
        unsigned gn = (unsigned)(nrg * (TILE * 4)) + (unsigned)(tid * 16);
        async_tile_load(ldsn, gn, msg);
        WAIT_ASYNCCNT(2);   // the 2 older copies (current buffer) are complete
      } else {
        WAIT_ASYNCCNT(0);
      }
      __syncthreads();      // LDS written by other waves' async copies
    }

    const int4 c0 = *(const int4*)(vcol + (size_t)m * DEGc);
    const int4 c1 = *(const int4*)(vcol + (size_t)m * DEGc + 4);
    int col[8] = {c0.x, c0.y, c0.z, c0.w, c1.x, c1.y, c1.z, c1.w};

    float m1 = __builtin_inff(), m2 = __builtin_inff();
    int idx = 0;
    unsigned sb = 0;
#pragma unroll
    for (int d = 0; d < 8; ++d) {
      float mg = first ? 0.0f : smsg[buf][(mi * DEGc + d) * Bc + b];
      float lvv = lv[col[d] * Bc + b];      // 64 lanes -> 256B contiguous burst
      float av = lvv - mg;
      if (av < 0.0f) sb |= (1u << d);       // sign(+-0) treated as +1, per reference
      float mag = fabsf(av);
      if (mag < m1) { m2 = m1; m1 = mag; idx = d; }
      else if (mag < m2) { m2 = mag; }
    }
    const unsigned par = __popc(sb) & 1u;
    const float t = ALPHAc * sgnS[(size_t)m * Bc + b];
#pragma unroll
    for (int d = 0; d < 8; ++d) {
      float mag_ex = (d == idx) ? m2 : m1;
      unsigned sd = (sb >> d) & 1u;
      float r = (((par ^ sd) != 0u) ? -t : t) * mag_ex;
      if (col[d] == Nc) r = 0.0f;           // dummy edge -> zero message
      msg[(size_t)(m * DEGc + d) * Bc + b] = r;  // coalesced along b
    }
    if (!first) __syncthreads();  // all waves done reading smsg[buf] before reuse
    buf ^= 1;
  }
}

// Variable-centric gather: l_v[v,b] = u_init[v,b] + sum over CSR edges of col v.
// Deterministic (fixed CSR order), atomic-free, coalesced along b.
__global__ __launch_bounds__(TPB) void k_gather(const float* __restrict__ U,
                                                const float* __restrict__ msg,
                                                const int* __restrict__ rowptr,
                                                const int* __restrict__ edges,
                                                float* __restrict__ L) {
  int i = blockIdx.x * TPB + threadIdx.x;
  if (i >= (Nc + 1) * Bc) return;
  int b = i & 63, v = i >> 6;
  float acc = U[i];
  int s = rowptr[v], e = rowptr[v + 1];
  for (int k = s; k < e; ++k) {
    int eid = edges[k];                       // uniform across the 64-lane group
    acc += msg[(size_t)eid * Bc + b];         // 256B contiguous burst
  }
  L[i] = acc;
}

// transpose [N+1,B] -> [B,N+1]
__global__ __launch_bounds__(TPB) void k_out(const float* __restrict__ L,
                                             float* __restrict__ out) {
  int i = blockIdx.x * TPB + threadIdx.x;
  if (i >= Bc * (Nc + 1)) return;
  int b = i / (Nc + 1);
  int v = i - b * (Nc + 1);
  out[i] = L[(size_t)v * Bc + b];
}

extern "C" void kernel_launch(void* const* d_in, const int* in_sizes, int n_in,
                              void* d_out, int out_size, void* d_ws, size_t ws_size,
                              hipStream_t stream) {
  const float* llr0 = (const float*)d_in[0];   // [B, N]
  const int* synd   = (const int*)d_in[1];     // [B, M]
  const int* vcol   = (const int*)d_in[2];     // [M, DEG]
  float* out = (float*)d_out;                  // [B, N+1]

  // workspace layout (~27.7 MB total)
  float* U   = (float*)d_ws;                               // (N+1)*B
  float* L   = U + (size_t)(Nc + 1) * Bc;                  // (N+1)*B
  float* S   = L + (size_t)(Nc + 1) * Bc;                  // M*B
  float* msg = S + (size_t)Mc * Bc;                        // M*DEG*B (16B aligned)
  int* counts = (int*)(msg + (size_t)Mc * DEGc * Bc);      // N+1
  int* rowptr = counts + (Nc + 1);                         // N+2
  int* cursor = rowptr + (Nc + 2);                         // N+1
  int* edges  = cursor + (Nc + 1);                         // M*DEG

  const int NVB = (Nc + 1) * Bc;      // 1,048,640
  const int NE  = Mc * DEGc;          // 65,536

  k_build_u<<<(NVB + TPB - 1) / TPB, TPB, 0, stream>>>(llr0, U);
  k_sgns<<<(Mc * Bc + TPB - 1) / TPB, TPB, 0, stream>>>(synd, S);
  k_zero_i32<<<(Nc + 1 + TPB - 1) / TPB, TPB, 0, stream>>>(counts, Nc + 1);
  k_count<<<(NE + TPB - 1) / TPB, TPB, 0, stream>>>(vcol, counts);
  k_scan<<<1, TPB, 0, stream>>>(counts, rowptr);
  k_cursor<<<(Nc + 1 + TPB - 1) / TPB, TPB, 0, stream>>>(rowptr, cursor);
  k_fill<<<(NE + TPB - 1) / TPB, TPB, 0, stream>>>(vcol, cursor, edges);
  k_sort<<<(Nc + 1 + TPB - 1) / TPB, TPB, 0, stream>>>(rowptr, edges);

  for (int it = 0; it < ITERS; ++it) {
    k_check<<<CHK_BLOCKS, TPB, 0, stream>>>(it == 0 ? U : L, msg, S, vcol,
                                            it == 0 ? 1 : 0);
    k_gather<<<(NVB + TPB - 1) / TPB, TPB, 0, stream>>>(U, msg, rowptr, edges, L);
  }

  k_out<<<(Bc * (Nc + 1) + TPB - 1) / TPB, TPB, 0, stream>>>(L, out);
}